// TransformerBlockQuantum_65481071406855
// MI455X (gfx1250) — compile-verified
//
#include <hip/hip_runtime.h>
#include <math.h>

#define EMBED 8
#define HEADS 4
#define DK    2
#define FFN_  1024
#define SEQ   2048
#define BATCH 8
#define TOKENS (BATCH*SEQ)
#define EPS 1e-5f

typedef float v2f __attribute__((ext_vector_type(2)));
typedef float v8f __attribute__((ext_vector_type(8)));

// ---------------------------------------------------------------------------
// Kernel 1: quantum projections q,k,v = cumprod(cos(x + theta)) per token.
// q/k/v are written HEAD-MAJOR: (B, H, S, DK), so the attention kernel's
// per-chunk K/V loads are 16 consecutive 8B pairs = one 128B line per half.
// ---------------------------------------------------------------------------
__global__ void qkv_kernel(const float* __restrict__ x,
                           const float* __restrict__ thq,
                           const float* __restrict__ thk,
                           const float* __restrict__ thv,
                           float* __restrict__ q,
                           float* __restrict__ k,
                           float* __restrict__ v) {
    int t = blockIdx.x * blockDim.x + threadIdx.x;
    if (t >= TOKENS) return;
    const int b = t / SEQ;
    const int s = t - b * SEQ;

    const float* xp = x + (size_t)t * EMBED;
    float xv[EMBED];
#pragma unroll
    for (int e = 0; e < EMBED; ++e) xv[e] = xp[e];

    float cq[EMBED], ck[EMBED], cv[EMBED];
    float pq = 1.f, pk = 1.f, pv = 1.f;
#pragma unroll
    for (int e = 0; e < EMBED; ++e) {
        pq *= cosf(xv[e] + thq[e]);  cq[e] = pq;
        pk *= cosf(xv[e] + thk[e]);  ck[e] = pk;
        pv *= cosf(xv[e] + thv[e]);  cv[e] = pv;
    }

#pragma unroll
    for (int h = 0; h < HEADS; ++h) {
        const size_t off = ((size_t)(b * HEADS + h) * SEQ + s) * DK;
        v2f r;
        r.x = cq[h * DK]; r.y = cq[h * DK + 1]; *(v2f*)(q + off) = r;
        r.x = ck[h * DK]; r.y = ck[h * DK + 1]; *(v2f*)(k + off) = r;
        r.x = cv[h * DK]; r.y = cv[h * DK + 1]; *(v2f*)(v + off) = r;
    }
}

// ---------------------------------------------------------------------------
// Kernel 2: attention. One wave = 16 queries of one (b,h).
// Scores via V_WMMA_F32_16X16X4_F32 (DK=2 padded to K=4).
//  * |q_i|,|k_i| <= 1 (cumprod of cosines) => |score| <= sqrt(2): softmax
//    needs NO max subtraction (exp can't overflow/underflow).
//  * A is pre-scaled by log2(e)/sqrt(2): WMMA emits log2-domain scores, so
//    the softmax exp is a bare v_exp_f32 (2^x) -- no per-chunk muls.
//  * A's lanes 16-31 (K=2,3) are zero, so B's lanes 16-31 are don't-care:
//    the raw K-pair load is passed unmasked => fully branchless inner loop.
//  * Head-major K/V layout => each chunk load is one contiguous 128B line.
// One 16-lane shfl reduction at the end. No LDS in the hot loop.
// ---------------------------------------------------------------------------
__global__ __launch_bounds__(256) void attn_kernel(const float* __restrict__ q,
                                                   const float* __restrict__ k,
                                                   const float* __restrict__ v,
                                                   float* __restrict__ o) {
    const int wave = (blockIdx.x * blockDim.x + threadIdx.x) >> 5;
    const int lane = threadIdx.x & 31;
    const int half = lane >> 4;      // 0: rows 0-7, 1: rows 8-15 of C tile
    const int col  = lane & 15;      // N index within tile

    const int qt = wave & 127;            // S/16 = 128 query tiles
    const int h  = (wave >> 7) & (HEADS - 1);
    const int b  = wave >> 9;
    const int qbase = qt * 16;

    const size_t head_off = (size_t)(b * HEADS + h) * SEQ * DK;

    // A operand (Q tile, 16x4): lanes 0-15 hold M=lane, K=0..1; lanes 16-31
    // hold K=2..3 which must be zero (this also makes B's lanes 16-31 moot).
    v2f a;
    if (half == 0) {
        const v2f qv = *(const v2f*)(q + head_off + (size_t)(qbase + lane) * DK);
        const float scale = 1.02013880099f;   // log2(e) / sqrt(DK)
        a.x = qv.x * scale;
        a.y = qv.y * scale;
    } else {
        a.x = 0.f; a.y = 0.f;
    }

    // Per-lane partial sums for 8 rows (this half's rows j + 8*half).
    float l[8], a0[8], a1[8];
#pragma unroll
    for (int j = 0; j < 8; ++j) { l[j] = 0.f; a0[j] = 0.f; a1[j] = 0.f; }

    const float* kbase = k + head_off + (size_t)col * DK;
    const float* vbase = v + head_off + (size_t)col * DK;

#pragma unroll 4
    for (int kc = 0; kc < SEQ / 16; ++kc) {
        // B operand (K^T tile, 4x16): rows K=0,1 in lanes 0-15 (VGPR0/1);
        // lanes 16-31 (rows K=2,3) carry duplicate finite data that is
        // multiplied by A's zeros -- no masking needed.
        const v2f bb = *(const v2f*)(kbase + (size_t)kc * 16 * DK);
        const v2f vv = *(const v2f*)(vbase + (size_t)kc * 16 * DK);
        const float v0 = vv.x;
        const float v1 = vv.y;

        v8f c = {};
        c = __builtin_amdgcn_wmma_f32_16x16x4_f32(
                /*neg_a=*/false, a, /*neg_b=*/false, bb,
                /*c_mod=*/(short)0, c, /*reuse_a=*/false, /*reuse_b=*/false);

        // Unnormalized softmax accumulation, log2 domain: p = 2^c = exp(score).
#pragma unroll
        for (int j = 0; j < 8; ++j) {
            const float p = __builtin_amdgcn_exp2f(c[j]);
            l[j]  += p;
            a0[j] += p * v0;
            a1[j] += p * v1;
        }
    }

    // Reduce per-column partials over the 16 lanes of each half, then write.
    // Output layout is token-major (B,S,E) for the tail kernel.
#pragma unroll
    for (int j = 0; j < 8; ++j) {
        float lj = l[j], x0 = a0[j], x1 = a1[j];
        lj += __shfl_xor(lj, 1); x0 += __shfl_xor(x0, 1); x1 += __shfl_xor(x1, 1);
        lj += __shfl_xor(lj, 2); x0 += __shfl_xor(x0, 2); x1 += __shfl_xor(x1, 2);
        lj += __shfl_xor(lj, 4); x0 += __shfl_xor(x0, 4); x1 += __shfl_xor(x1, 4);
        lj += __shfl_xor(lj, 8); x0 += __shfl_xor(x0, 8); x1 += __shfl_xor(x1, 8);
        if (col == j) {
            const int row = qbase + j + half * 8;
            const float inv = 1.f / lj;
            v2f* op = (v2f*)(o + ((size_t)(b * SEQ + row) * EMBED) + h * DK);
            v2f r; r.x = x0 * inv; r.y = x1 * inv;
            *op = r;
        }
    }
}

// ---------------------------------------------------------------------------
// Kernel 3: attn@Wc + bc, residual, LN1, qproj(theta_f), FFN, residual, LN2.
// One thread per token; weight loads are thread-uniform (scalar broadcast).
// ---------------------------------------------------------------------------
__global__ void tail_kernel(const float* __restrict__ x,
                            const float* __restrict__ ao,
                            const float* __restrict__ Wc,
                            const float* __restrict__ bc,
                            const float* __restrict__ g1,
                            const float* __restrict__ bb1,
                            const float* __restrict__ thf,
                            const float* __restrict__ W1,
                            const float* __restrict__ bf1,
                            const float* __restrict__ W2,
                            const float* __restrict__ bf2,
                            const float* __restrict__ g2,
                            const float* __restrict__ bb2,
                            float* __restrict__ out) {
    int t = blockIdx.x * blockDim.x + threadIdx.x;
    if (t >= TOKENS) return;

    float xv[8], av[8];
#pragma unroll
    for (int e = 0; e < 8; ++e) { xv[e] = x[(size_t)t * 8 + e]; av[e] = ao[(size_t)t * 8 + e]; }

    // attn output projection + residual
    float h1[8];
#pragma unroll
    for (int e = 0; e < 8; ++e) {
        float acc = bc[e];
#pragma unroll
        for (int j = 0; j < 8; ++j) acc += av[j] * Wc[j * 8 + e];
        h1[e] = xv[e] + acc;
    }

    // LayerNorm 1
    float mean = 0.f;
#pragma unroll
    for (int e = 0; e < 8; ++e) mean += h1[e];
    mean *= 0.125f;
    float var = 0.f;
#pragma unroll
    for (int e = 0; e < 8; ++e) { float d = h1[e] - mean; var += d * d; }
    var *= 0.125f;
    float r = rsqrtf(var + EPS);
    float x1[8];
#pragma unroll
    for (int e = 0; e < 8; ++e) x1[e] = (h1[e] - mean) * r * g1[e] + bb1[e];

    // quantum feed-forward projection
    float fv[8];
    {
        float c = 1.f;
#pragma unroll
        for (int e = 0; e < 8; ++e) { c *= cosf(x1[e] + thf[e]); fv[e] = c; }
    }

    // FFN: relu(f@W1 + b1) @ W2 + b2
    float o8[8];
#pragma unroll
    for (int e = 0; e < 8; ++e) o8[e] = bf2[e];
    for (int ff = 0; ff < FFN_; ++ff) {
        float acc = bf1[ff];
#pragma unroll
        for (int e = 0; e < 8; ++e) acc += fv[e] * W1[e * FFN_ + ff];
        acc = fmaxf(acc, 0.f);
#pragma unroll
        for (int e = 0; e < 8; ++e) o8[e] += acc * W2[ff * 8 + e];
    }

    // residual + LayerNorm 2
    float h2[8];
#pragma unroll
    for (int e = 0; e < 8; ++e) h2[e] = x1[e] + o8[e];
    float mean2 = 0.f;
#pragma unroll
    for (int e = 0; e < 8; ++e) mean2 += h2[e];
    mean2 *= 0.125f;
    float var2 = 0.f;
#pragma unroll
    for (int e = 0; e < 8; ++e) { float d = h2[e] - mean2; var2 += d * d; }
    var2 *= 0.125f;
    float r2 = rsqrtf(var2 + EPS);
#pragma unroll
    for (int e = 0; e < 8; ++e)
        out[(size_t)t * 8 + e] = (h2[e] - mean2) * r2 * g2[e] + bb2[e];
}

// ---------------------------------------------------------------------------
extern "C" void kernel_launch(void* const* d_in, const int* in_sizes, int n_in,
                              void* d_out, int out_size, void* d_ws, size_t ws_size,
                              hipStream_t stream) {
    const float* x     = (const float*)d_in[0];
    const float* thq   = (const float*)d_in[1];
    const float* thk   = (const float*)d_in[2];
    const float* thv   = (const float*)d_in[3];
    const float* Wc    = (const float*)d_in[4];
    const float* bc    = (const float*)d_in[5];
    const float* ln1_g = (const float*)d_in[6];
    const float* ln1_b = (const float*)d_in[7];
    const float* thf   = (const float*)d_in[8];
    const float* W1    = (const float*)d_in[9];
    const float* b1    = (const float*)d_in[10];
    const float* W2    = (const float*)d_in[11];
    const float* b2    = (const float*)d_in[12];
    const float* ln2_g = (const float*)d_in[13];
    const float* ln2_b = (const float*)d_in[14];
    float* out = (float*)d_out;

    float* ws = (float*)d_ws;
    float* q  = ws;                         // TOKENS*EMBED = 131072 floats each
    float* k  = ws + (size_t)TOKENS * EMBED;
    float* v  = ws + (size_t)2 * TOKENS * EMBED;
    float* ao = ws + (size_t)3 * TOKENS * EMBED;

    qkv_kernel<<<TOKENS / 256, 256, 0, stream>>>(x, thq, thk, thv, q, k, v);

    // 4096 waves = BATCH*HEADS*(SEQ/16); 8 waves per 256-thread block
    attn_kernel<<<(BATCH * HEADS * (SEQ / 16)) / 8, 256, 0, stream>>>(q, k, v, ao);

    tail_kernel<<<TOKENS / 256, 256, 0, stream>>>(x, ao, Wc, bc, ln1_g, ln1_b,
                                                  thf, W1, b1, W2, b2,
                                                  ln2_g, ln2_b, out);
}